// Mamba_1_64716567216492
// MI455X (gfx1250) — compile-verified
//
#include <hip/hip_runtime.h>
#include <hip/hip_bf16.h>

// ---------------------------------------------------------------------------
// Model constants
#define BB      32
#define C_IN    200
#define DIM     512
#define DEPTH   2
#define D_INNER 1024
#define D_STATE 16
#define D_CONV  4
#define DT_RANK 32          // (512+15)/16
#define L_TOK   128
#define NPATCH  256         // H*W
#define NCLS    16

typedef float v2f __attribute__((ext_vector_type(2)));
typedef float v8f __attribute__((ext_vector_type(8)));
typedef int   v2i_t __attribute__((vector_size(8)));    // async b64 operand type
typedef int   v4i_t __attribute__((vector_size(16)));   // async b128 operand type

#define GLOBAL_AS __attribute__((address_space(1)))
#define LDS_AS    __attribute__((address_space(3)))

#define WMMA_F32(a, b, c) \
  __builtin_amdgcn_wmma_f32_16x16x4_f32(false, (a), false, (b), (short)0, (c), false, false)

// CDNA5 async global->LDS copies (ASYNCcnt), with graceful fallback.
#if defined(__gfx1250__) && \
    __has_builtin(__builtin_amdgcn_global_load_async_to_lds_b64) && \
    __has_builtin(__builtin_amdgcn_global_load_async_to_lds_b128) && \
    __has_builtin(__builtin_amdgcn_s_wait_asynccnt)
#define USE_ASYNC_LDS 1
#else
#define USE_ASYNC_LDS 0
#endif

// ---------------------------------------------------------------------------
// Store one 16x16 f32 C/D fragment with fused bias / residual / activation.
// Lane layout: col = lane&15, rows = (lane>>4)*8 + i.
__device__ __forceinline__ void store_tile(const v8f& acc, float* __restrict__ C, int ldc,
                                           int rbase, int c, int M, int N,
                                           const float* __restrict__ bias,
                                           const float* __restrict__ resid, int ldr,
                                           int act) {
  if (c >= N) return;
  const float bval = bias ? bias[c] : 0.0f;
#pragma unroll
  for (int i = 0; i < 8; ++i) {
    const int r = rbase + i;
    if (r < M) {
      float v = acc[i] + bval;
      if (resid) v += resid[(long long)r * ldr + c];
      if (act == 1)      v = fmaxf(v, 0.0f);                         // relu
      else if (act == 2) v = (v > 20.0f) ? v : __logf(1.0f + __expf(v)); // softplus
      C[(long long)r * ldc + c] = v;
    }
  }
}

// ---------------------------------------------------------------------------
// Direct-from-global f32 WMMA GEMM, 32x32 per wave (2x2 accumulators).
// Used for irregular shapes (odd K / transposed / batched / tiny M).
// A accessed as A[m*rsA + k*csA], B as B[k*rsB + n*csB] (general strides).
// AC: csA==1 (A fragment contiguous -> b64 load). BC: rsB==1 (same for B).
// Requires K % 4 == 0.
template <bool AC, bool BC>
__global__ void gemm32_wmma(const float* __restrict__ A, int rsA, int csA, long long batA,
                            const float* __restrict__ B, int rsB, int csB, long long batB,
                            float* __restrict__ C, int ldc, long long batC,
                            const float* __restrict__ bias,
                            const float* __restrict__ resid, int ldr, long long batR,
                            int M, int N, int K, int act) {
  A += blockIdx.z * batA;
  B += blockIdx.z * batB;
  C += blockIdx.z * batC;
  if (resid) resid += blockIdx.z * batR;

  const int lane = threadIdx.x;
  const int l15 = lane & 15;
  const int half = lane >> 4;
  const int kb = half * 2;
  const int m0 = blockIdx.y << 5, n0 = blockIdx.x << 5;

  const int ar0 = m0 + l15, ar1 = ar0 + 16;
  const int bc0 = n0 + l15, bc1 = bc0 + 16;

  const float* pa0 = A + (long long)min(ar0, M - 1) * rsA + (long long)kb * csA;
  const float* pa1 = A + (long long)min(ar1, M - 1) * rsA + (long long)kb * csA;
  const float* pb0 = B + (long long)min(bc0, N - 1) * csB + (long long)kb * rsB;
  const float* pb1 = B + (long long)min(bc1, N - 1) * csB + (long long)kb * rsB;
  const long long stepA = 4LL * csA;
  const long long stepB = 4LL * rsB;

  v8f a00, a01, a10, a11;
#pragma unroll
  for (int i = 0; i < 8; ++i) { a00[i] = 0.f; a01[i] = 0.f; a10[i] = 0.f; a11[i] = 0.f; }

  const bool interior = (m0 + 32 <= M) && (n0 + 32 <= N);

  if (interior) {
#pragma unroll 2
    for (int k = 0; k < K; k += 4) {
      v2f fa0, fa1, fb0, fb1;
      if (AC) { fa0 = *(const v2f*)pa0; fa1 = *(const v2f*)pa1; }
      else    { fa0.x = pa0[0]; fa0.y = pa0[csA]; fa1.x = pa1[0]; fa1.y = pa1[csA]; }
      if (BC) { fb0 = *(const v2f*)pb0; fb1 = *(const v2f*)pb1; }
      else    { fb0.x = pb0[0]; fb0.y = pb0[rsB]; fb1.x = pb1[0]; fb1.y = pb1[rsB]; }
      a00 = WMMA_F32(fa0, fb0, a00);
      a01 = WMMA_F32(fa0, fb1, a01);
      a10 = WMMA_F32(fa1, fb0, a10);
      a11 = WMMA_F32(fa1, fb1, a11);
      pa0 += stepA; pa1 += stepA; pb0 += stepB; pb1 += stepB;
    }
  } else {
    const float mA0 = (ar0 < M) ? 1.f : 0.f;
    const float mA1 = (ar1 < M) ? 1.f : 0.f;
    const float mB0 = (bc0 < N) ? 1.f : 0.f;
    const float mB1 = (bc1 < N) ? 1.f : 0.f;
    for (int k = 0; k < K; k += 4) {
      v2f fa0, fa1, fb0, fb1;
      if (AC) { fa0 = *(const v2f*)pa0; fa1 = *(const v2f*)pa1; }
      else    { fa0.x = pa0[0]; fa0.y = pa0[csA]; fa1.x = pa1[0]; fa1.y = pa1[csA]; }
      if (BC) { fb0 = *(const v2f*)pb0; fb1 = *(const v2f*)pb1; }
      else    { fb0.x = pb0[0]; fb0.y = pb0[rsB]; fb1.x = pb1[0]; fb1.y = pb1[rsB]; }
      fa0 *= mA0; fa1 *= mA1; fb0 *= mB0; fb1 *= mB1;
      a00 = WMMA_F32(fa0, fb0, a00);
      a01 = WMMA_F32(fa0, fb1, a01);
      a10 = WMMA_F32(fa1, fb0, a10);
      a11 = WMMA_F32(fa1, fb1, a11);
      pa0 += stepA; pa1 += stepA; pb0 += stepB; pb1 += stepB;
    }
  }

  const int rb0 = m0 + half * 8;
  store_tile(a00, C, ldc, rb0,      n0 + l15,      M, N, bias, resid, ldr, act);
  store_tile(a01, C, ldc, rb0,      n0 + 16 + l15, M, N, bias, resid, ldr, act);
  store_tile(a10, C, ldc, rb0 + 16, n0 + l15,      M, N, bias, resid, ldr, act);
  store_tile(a11, C, ldc, rb0 + 16, n0 + 16 + l15, M, N, bias, resid, ldr, act);
}

// ---------------------------------------------------------------------------
// LDS-staged, double-buffered f32 WMMA GEMM for the heavy regular GEMMs.
// Block = 128 threads (2x2 waves), 64x64 tile, K-step 16.
// Requires: A row-major (csA==1, lda=rsA), B row-major (csB==1, ldb=rsB),
//           M%64==0, N%64==0, K%16==0.
// sA[buf][m][k] padded to 18 (even pad -> conflict-free b64 fragment loads).
// sB[buf][k][n] K-major (conflict-free b32 fragment loads, b128 fills).
// Async path: 6 GLOBAL_LOAD_ASYNC_TO_LDS ops/wave/tile (4x b64 A + 2x b128 B),
// double-buffered with s_wait_asynccnt(6)/(0).
__global__ void gemm64_lds_wmma(const float* __restrict__ A, int lda,
                                const float* __restrict__ B, int ldb,
                                float* __restrict__ C, int ldc,
                                const float* __restrict__ bias,
                                const float* __restrict__ resid, int ldr,
                                int M, int N, int K, int act) {
  __shared__ float sA[2][64][18];
  __shared__ float sB[2][16][64];

  const int tid = threadIdx.x;
  const int wave = tid >> 5;
  const int lane = tid & 31;
  const int l15 = lane & 15;
  const int half = lane >> 4;
  const int kb = half * 2;
  const int wm = (wave >> 1) * 32;   // wave quadrant inside 64x64
  const int wn = (wave & 1) * 32;
  const int m0 = blockIdx.y << 6, n0 = blockIdx.x << 6;

  // global -> LDS staging map
  const int arow = tid >> 1;               // 0..63
  const int ak0  = (tid & 1) * 8;          // 0 or 8
  const int brow = tid & 15;               // 0..15
  const int bn0  = (tid >> 4) * 8;         // 0..56

  const float* gA = A + (long long)(m0 + arow) * lda + ak0;
  const float* gB = B + (long long)brow * ldb + n0 + bn0;

  v8f a00, a01, a10, a11;
#pragma unroll
  for (int i = 0; i < 8; ++i) { a00[i] = 0.f; a01[i] = 0.f; a10[i] = 0.f; a11[i] = 0.f; }

#if USE_ASYNC_LDS
  // ---- async double-buffered pipeline -------------------------------------
  auto issue = [&](int kt, int buf) {
#pragma unroll
    for (int j = 0; j < 4; ++j)
      __builtin_amdgcn_global_load_async_to_lds_b64(
          (GLOBAL_AS v2i_t*)(gA + kt + 2 * j),
          (LDS_AS v2i_t*)&sA[buf][arow][ak0 + 2 * j], 0, 0);
#pragma unroll
    for (int j = 0; j < 2; ++j)
      __builtin_amdgcn_global_load_async_to_lds_b128(
          (GLOBAL_AS v4i_t*)(gB + (long long)kt * ldb + 4 * j),
          (LDS_AS v4i_t*)&sB[buf][brow][bn0 + 4 * j], 0, 0);
  };

  int buf = 0;
  issue(0, 0);
  for (int kt = 0; kt < K; kt += 16) {
    const bool has_next = (kt + 16) < K;
    if (has_next) issue(kt + 16, buf ^ 1);
    if (has_next) __builtin_amdgcn_s_wait_asynccnt(6);
    else          __builtin_amdgcn_s_wait_asynccnt(0);
    __syncthreads();                              // tile[buf] visible to all waves
#pragma unroll
    for (int kk = 0; kk < 16; kk += 4) {
      v2f fa0 = *(const v2f*)&sA[buf][wm + l15][kk + kb];
      v2f fa1 = *(const v2f*)&sA[buf][wm + 16 + l15][kk + kb];
      v2f fb0, fb1;
      fb0.x = sB[buf][kk + kb][wn + l15];      fb0.y = sB[buf][kk + kb + 1][wn + l15];
      fb1.x = sB[buf][kk + kb][wn + 16 + l15]; fb1.y = sB[buf][kk + kb + 1][wn + 16 + l15];
      a00 = WMMA_F32(fa0, fb0, a00);
      a01 = WMMA_F32(fa0, fb1, a01);
      a10 = WMMA_F32(fa1, fb0, a10);
      a11 = WMMA_F32(fa1, fb1, a11);
    }
    __syncthreads();                              // done reading buf before re-issue
    buf ^= 1;
  }
#else
  // ---- fallback: register-staged double-buffered pipeline -----------------
  v2f ra[4];
  float4 rb[2];
  auto stage_load = [&](int kt) {
#pragma unroll
    for (int j = 0; j < 4; ++j) ra[j] = *(const v2f*)(gA + kt + 2 * j);
#pragma unroll
    for (int j = 0; j < 2; ++j) rb[j] = *(const float4*)(gB + (long long)kt * ldb + 4 * j);
  };
  auto stage_store = [&](int buf) {
#pragma unroll
    for (int j = 0; j < 4; ++j) *(v2f*)&sA[buf][arow][ak0 + 2 * j] = ra[j];
#pragma unroll
    for (int j = 0; j < 2; ++j) *(float4*)&sB[buf][brow][bn0 + 4 * j] = rb[j];
  };

  int buf = 0;
  stage_load(0);
  stage_store(0);
  __syncthreads();
  for (int kt = 0; kt < K; kt += 16) {
    const bool has_next = (kt + 16) < K;
    if (has_next) { stage_load(kt + 16); stage_store(buf ^ 1); }
#pragma unroll
    for (int kk = 0; kk < 16; kk += 4) {
      v2f fa0 = *(const v2f*)&sA[buf][wm + l15][kk + kb];
      v2f fa1 = *(const v2f*)&sA[buf][wm + 16 + l15][kk + kb];
      v2f fb0, fb1;
      fb0.x = sB[buf][kk + kb][wn + l15];      fb0.y = sB[buf][kk + kb + 1][wn + l15];
      fb1.x = sB[buf][kk + kb][wn + 16 + l15]; fb1.y = sB[buf][kk + kb + 1][wn + 16 + l15];
      a00 = WMMA_F32(fa0, fb0, a00);
      a01 = WMMA_F32(fa0, fb1, a01);
      a10 = WMMA_F32(fa1, fb0, a10);
      a11 = WMMA_F32(fa1, fb1, a11);
    }
    __syncthreads();
    buf ^= 1;
  }
#endif

  const int rb0 = m0 + wm + half * 8;
  const int cc0 = n0 + wn + l15;
  store_tile(a00, C, ldc, rb0,      cc0,      M, N, bias, resid, ldr, act);
  store_tile(a01, C, ldc, rb0,      cc0 + 16, M, N, bias, resid, ldr, act);
  store_tile(a10, C, ldc, rb0 + 16, cc0,      M, N, bias, resid, ldr, act);
  store_tile(a11, C, ldc, rb0 + 16, cc0 + 16, M, N, bias, resid, ldr, act);
}

// ---------------------------------------------------------------------------
// Snake flatten + LayerNorm(C_IN).  One block per (b, patch); 256 threads, t = channel.
__global__ void prep_tokens(const float* __restrict__ img,
                            const float* __restrict__ g, const float* __restrict__ b,
                            float* __restrict__ xln) {
  __shared__ float red[256];
  const int bid = blockIdx.x;            // b*256 + p
  const int bi = bid >> 8, p = bid & 255;
  const int r = p >> 4, c0 = p & 15;
  const int c = (r & 1) ? (15 - c0) : c0;
  const int t = threadIdx.x;

  float v = 0.0f;
  if (t < C_IN) v = img[((long long)bi * C_IN + t) * NPATCH + r * 16 + c];

  red[t] = v; __syncthreads();
  for (int s = 128; s > 0; s >>= 1) { if (t < s) red[t] += red[t + s]; __syncthreads(); }
  const float mean = red[0] / (float)C_IN; __syncthreads();

  float d = (t < C_IN) ? (v - mean) : 0.0f;
  red[t] = d * d; __syncthreads();
  for (int s = 128; s > 0; s >>= 1) { if (t < s) red[t] += red[t + s]; __syncthreads(); }
  const float var = red[0] / (float)C_IN;

  if (t < C_IN)
    xln[(long long)bid * C_IN + t] = (v - mean) * rsqrtf(var + 1e-5f) * g[t] + b[t];
}

// ---------------------------------------------------------------------------
// In-place LayerNorm(DIM) + sincos-2d pos-emb.  One block per row (b*256+p); 512 thr.
__global__ void ln_posemb(float* __restrict__ X,
                          const float* __restrict__ g, const float* __restrict__ b) {
  __shared__ float red[512];
  const int row = blockIdx.x;
  const int p = row & 255;
  const int d = threadIdx.x;

  const float v = X[(long long)row * DIM + d];
  red[d] = v; __syncthreads();
  for (int s = 256; s > 0; s >>= 1) { if (d < s) red[d] += red[d + s]; __syncthreads(); }
  const float mean = red[0] / (float)DIM; __syncthreads();
  const float df = v - mean;
  red[d] = df * df; __syncthreads();
  for (int s = 256; s > 0; s >>= 1) { if (d < s) red[d] += red[d + s]; __syncthreads(); }
  const float var = red[0] / (float)DIM;

  const float ln = df * rsqrtf(var + 1e-5f) * g[d] + b[d];
  const int q = d >> 7, i = d & 127;
  const float omega = __powf(10000.0f, -(float)i / 127.0f);
  const float coord = (q < 2) ? (float)(p & 15) : (float)(p >> 4); // x then y
  const float arg = coord * omega;
  const float pe = (q & 1) ? __cosf(arg) : __sinf(arg);
  X[(long long)row * DIM + d] = ln + pe;
}

// ---------------------------------------------------------------------------
// Softmax over the 256 patches for each (b, l).  S layout: (b*256 + p)*128 + l.
__global__ void softmax_p(float* __restrict__ S) {
  __shared__ float red[256];
  const int l = blockIdx.x, bi = blockIdx.y;
  const int p = threadIdx.x;
  const long long idx = ((long long)bi * NPATCH + p) * L_TOK + l;
  const float v = S[idx];

  red[p] = v; __syncthreads();
  for (int s = 128; s > 0; s >>= 1) { if (p < s) red[p] = fmaxf(red[p], red[p + s]); __syncthreads(); }
  const float mx = red[0]; __syncthreads();
  const float e = __expf(v - mx);
  red[p] = e; __syncthreads();
  for (int s = 128; s > 0; s >>= 1) { if (p < s) red[p] += red[p + s]; __syncthreads(); }
  S[idx] = e / red[0];
}

// ---------------------------------------------------------------------------
// RMSNorm over DIM.  One block per row (4096 rows), 512 threads.
__global__ void rmsnorm(const float* __restrict__ X, const float* __restrict__ w,
                        float* __restrict__ Y) {
  __shared__ float red[512];
  const int row = blockIdx.x;
  const int d = threadIdx.x;
  const float v = X[(long long)row * DIM + d];
  red[d] = v * v; __syncthreads();
  for (int s = 256; s > 0; s >>= 1) { if (d < s) red[d] += red[d + s]; __syncthreads(); }
  const float ms = red[0] / (float)DIM;
  Y[(long long)row * DIM + d] = v * rsqrtf(ms + 1e-5f) * w[d];
}

// ---------------------------------------------------------------------------
// Depthwise causal conv (k=4, left pad 3) + bias + SiLU.
__global__ void conv_silu(const float* __restrict__ xz, const float* __restrict__ cw,
                          const float* __restrict__ cb, float* __restrict__ xcs) {
  const int idx = blockIdx.x * blockDim.x + threadIdx.x;   // over 4096*1024
  const int d = idx & (D_INNER - 1);
  const int row = idx >> 10;                                // b*128 + t
  const int t = row & (L_TOK - 1);
  float acc = cb[d];
#pragma unroll
  for (int j = 0; j < 4; ++j) {
    const int tt = t + j - 3;
    if (tt >= 0)
      acc += cw[d * 4 + j] * xz[(long long)(row + j - 3) * (2 * D_INNER) + d];
  }
  xcs[idx] = acc / (1.0f + __expf(-acc));                   // silu
}

// ---------------------------------------------------------------------------
// Selective scan: 32 blocks (batch) x 1024 threads (d).  16-float state/thread.
__global__ void sel_scan(const float* __restrict__ xz,    // (4096,2048): z in cols [1024,2048)
                         const float* __restrict__ xcs,   // (4096,1024)
                         const float* __restrict__ dbc,   // (4096,64): B at 32.., C at 48..
                         const float* __restrict__ delta, // (4096,1024), softplus applied
                         const float* __restrict__ A_log, // (1024,16) this layer
                         const float* __restrict__ D_p,   // (1024)
                         float* __restrict__ ybuf) {      // (4096,1024)
  const int bi = blockIdx.x;
  const int d = threadIdx.x;
  __shared__ float sB[D_STATE], sC[D_STATE];

  float Arow[D_STATE];
#pragma unroll
  for (int n = 0; n < D_STATE; ++n) Arow[n] = -__expf(A_log[d * D_STATE + n]);
  const float Dv = D_p[d];

  float h[D_STATE];
#pragma unroll
  for (int n = 0; n < D_STATE; ++n) h[n] = 0.0f;

  for (int t = 0; t < L_TOK; ++t) {
    const int row = bi * L_TOK + t;
    if (d < D_STATE)          sB[d]      = dbc[(long long)row * 64 + DT_RANK + d];
    else if (d < 2 * D_STATE) sC[d - 16] = dbc[(long long)row * 64 + DT_RANK + D_STATE + (d - 16)];
    __syncthreads();

    const float xv = xcs[(long long)row * D_INNER + d];
    const float dl = delta[(long long)row * D_INNER + d];
    float y = 0.0f;
#pragma unroll
    for (int n = 0; n < D_STATE; ++n) {
      h[n] = __expf(dl * Arow[n]) * h[n] + dl * sB[n] * xv;
      y += h[n] * sC[n];
    }
    const float zv = xz[(long long)row * (2 * D_INNER) + D_INNER + d];
    const float sz = zv / (1.0f + __expf(-zv));
    ybuf[(long long)row * D_INNER + d] = (y + xv * Dv) * sz;
    __syncthreads();
  }
}

// ---------------------------------------------------------------------------
// Mean-pool over 128 tokens.  32 blocks x 512 threads.
__global__ void pool_mean(const float* __restrict__ X2, float* __restrict__ pooled) {
  const int bi = blockIdx.x, d = threadIdx.x;
  float s = 0.0f;
  for (int t = 0; t < L_TOK; ++t)
    s += X2[((long long)bi * L_TOK + t) * DIM + d];
  pooled[(long long)bi * DIM + d] = s / (float)L_TOK;
}

// ---------------------------------------------------------------------------
static inline void launch_gemm(hipStream_t st,
                               const float* A, int rsA, int csA, long long batA,
                               const float* B, int rsB, int csB, long long batB,
                               float* C, int ldc, long long batC,
                               const float* bias,
                               const float* resid, int ldr, long long batR,
                               int M, int N, int K, int act, int batches) {
  if (batches == 1 && csA == 1 && csB == 1 &&
      (M & 63) == 0 && (N & 63) == 0 && (K & 15) == 0) {
    dim3 g(N / 64, M / 64, 1);
    gemm64_lds_wmma<<<g, 128, 0, st>>>(A, rsA, B, rsB, C, ldc, bias, resid, ldr,
                                       M, N, K, act);
    return;
  }
  dim3 g((N + 31) / 32, (M + 31) / 32, batches);
  const bool ac = (csA == 1), bc = (rsB == 1);
  if (ac && bc)
    gemm32_wmma<true, true><<<g, 32, 0, st>>>(A, rsA, csA, batA, B, rsB, csB, batB,
                                              C, ldc, batC, bias, resid, ldr, batR, M, N, K, act);
  else if (ac)
    gemm32_wmma<true, false><<<g, 32, 0, st>>>(A, rsA, csA, batA, B, rsB, csB, batB,
                                               C, ldc, batC, bias, resid, ldr, batR, M, N, K, act);
  else if (bc)
    gemm32_wmma<false, true><<<g, 32, 0, st>>>(A, rsA, csA, batA, B, rsB, csB, batB,
                                               C, ldc, batC, bias, resid, ldr, batR, M, N, K, act);
  else
    gemm32_wmma<false, false><<<g, 32, 0, st>>>(A, rsA, csA, batA, B, rsB, csB, batB,
                                                C, ldc, batC, bias, resid, ldr, batR, M, N, K, act);
}

extern "C" void kernel_launch(void* const* d_in, const int* in_sizes, int n_in,
                              void* d_out, int out_size, void* d_ws, size_t ws_size,
                              hipStream_t stream) {
  const float* img       = (const float*)d_in[0];
  const float* ln1_g     = (const float*)d_in[1];
  const float* ln1_b     = (const float*)d_in[2];
  const float* w_emb     = (const float*)d_in[3];
  const float* b_emb     = (const float*)d_in[4];
  const float* ln2_g     = (const float*)d_in[5];
  const float* ln2_b     = (const float*)d_in[6];
  const float* token_wA  = (const float*)d_in[7];   // (128,512)
  const float* token_wV  = (const float*)d_in[8];   // (512,512)
  const float* rms_w     = (const float*)d_in[9];
  const float* in_proj_w = (const float*)d_in[10];  // (2,512,2048)
  const float* conv_w    = (const float*)d_in[11];  // (2,1024,4)
  const float* conv_b    = (const float*)d_in[12];
  const float* x_proj_w  = (const float*)d_in[13];  // (2,1024,64)
  const float* dt_proj_w = (const float*)d_in[14];  // (2,32,1024)
  const float* dt_proj_b = (const float*)d_in[15];
  const float* A_log     = (const float*)d_in[16];  // (2,1024,16)
  const float* D_p       = (const float*)d_in[17];
  const float* out_proj_w= (const float*)d_in[18];  // (2,1024,512)
  const float* head_w1   = (const float*)d_in[19];
  const float* head_b1   = (const float*)d_in[20];
  const float* head_w2   = (const float*)d_in[21];
  const float* head_b2   = (const float*)d_in[22];
  const float* reg_w1    = (const float*)d_in[23];
  const float* reg_b1    = (const float*)d_in[24];
  const float* reg_w2    = (const float*)d_in[25];
  const float* reg_b2    = (const float*)d_in[26];
  float* out = (float*)d_out;

  // ---- workspace layout (floats). Peak ~102 MB; two phases alias one arena.
  float* ws    = (float*)d_ws;
  float* X2    = ws;                            // 4096*512, persistent (tokens / residual)
  float* arena = ws + (long long)4096 * 512;
  // phase 1
  float* xln = arena;                           // 8192*200
  float* Xe  = xln + (long long)8192 * 200;     // 8192*512 (embedded patches)
  float* S   = Xe + (long long)8192 * 512;      // 8192*128 (attn logits / softmax)
  float* VV  = S + (long long)8192 * 128;       // 8192*512
  // phase 2 (aliases phase-1 region)
  float* xn    = arena;                         // 4096*512
  float* xz    = xn + (long long)4096 * 512;    // 4096*2048
  float* xcs   = xz + (long long)4096 * 2048;   // 4096*1024
  float* dbc   = xcs + (long long)4096 * 1024;  // 4096*64
  float* delta = dbc + (long long)4096 * 64;    // 4096*1024
  float* ybuf  = delta + (long long)4096 * 1024;// 4096*1024
  float* pooled= ybuf + (long long)4096 * 1024; // 32*512
  float* h1    = pooled + 32 * 512;             // 32*512
  float* r1    = h1 + 32 * 512;                 // 32*512

  const int NP = BB * NPATCH;    // 8192 patch rows
  const int NT = BB * L_TOK;     // 4096 token rows

  // 1) snake flatten + LN1
  prep_tokens<<<NP, 256, 0, stream>>>(img, ln1_g, ln1_b, xln);
  // 2) embed GEMM: Xe = xln(8192x200) @ w_emb(200x512) + b_emb  (K=200 -> direct kernel)
  launch_gemm(stream, xln, C_IN, 1, 0, w_emb, DIM, 1, 0, Xe, DIM, 0,
              b_emb, nullptr, 0, 0, NP, DIM, C_IN, 0, 1);
  // 3) LN2 + posemb (in place)
  ln_posemb<<<NP, 512, 0, stream>>>(Xe, ln2_g, ln2_b);
  // 4) S[b,p,l] = Xe @ wA^T   (B transposed via strides -> direct kernel, rsB==1)
  launch_gemm(stream, Xe, DIM, 1, 0, token_wA, 1, DIM, 0, S, L_TOK, 0,
              nullptr, nullptr, 0, 0, NP, L_TOK, DIM, 0, 1);
  // 5) softmax over p for each (b,l)
  softmax_p<<<dim3(L_TOK, BB), 256, 0, stream>>>(S);
  // 6) VV = Xe @ token_wV   (regular -> LDS kernel)
  launch_gemm(stream, Xe, DIM, 1, 0, token_wV, DIM, 1, 0, VV, DIM, 0,
              nullptr, nullptr, 0, 0, NP, DIM, DIM, 0, 1);
  // 7) X2_b = softmax(S)_b^T (128x256) @ VV_b (256x512), batched over b (direct kernel).
  launch_gemm(stream, S, 1, L_TOK, (long long)NPATCH * L_TOK,
              VV, DIM, 1, (long long)NPATCH * DIM,
              X2, DIM, (long long)L_TOK * DIM,
              nullptr, nullptr, 0, 0, L_TOK, DIM, NPATCH, 0, BB);

  // ---- Mamba layers
  for (int i = 0; i < DEPTH; ++i) {
    const float* ipw = in_proj_w + (long long)i * DIM * 2 * D_INNER;
    const float* cw  = conv_w   + (long long)i * D_INNER * D_CONV;
    const float* cb  = conv_b   + (long long)i * D_INNER;
    const float* xpw = x_proj_w + (long long)i * D_INNER * 64;
    const float* dpw = dt_proj_w+ (long long)i * DT_RANK * D_INNER;
    const float* dpb = dt_proj_b+ (long long)i * D_INNER;
    const float* al  = A_log    + (long long)i * D_INNER * D_STATE;
    const float* dp  = D_p      + (long long)i * D_INNER;
    const float* opw = out_proj_w + (long long)i * D_INNER * DIM;

    rmsnorm<<<NT, 512, 0, stream>>>(X2, rms_w + (long long)i * DIM, xn);
    // xz = xn(4096x512) @ in_proj(512x2048)      (LDS kernel)
    launch_gemm(stream, xn, DIM, 1, 0, ipw, 2 * D_INNER, 1, 0, xz, 2 * D_INNER, 0,
                nullptr, nullptr, 0, 0, NT, 2 * D_INNER, DIM, 0, 1);
    conv_silu<<<(NT * D_INNER) / 256, 256, 0, stream>>>(xz, cw, cb, xcs);
    // dbc = xcs(4096x1024) @ x_proj(1024x64)     (LDS kernel)
    launch_gemm(stream, xcs, D_INNER, 1, 0, xpw, 64, 1, 0, dbc, 64, 0,
                nullptr, nullptr, 0, 0, NT, 64, D_INNER, 0, 1);
    // delta = softplus(dt @ dt_proj + b), dt = dbc[:, :32] (lda=64, K=32 -> LDS kernel)
    launch_gemm(stream, dbc, 64, 1, 0, dpw, D_INNER, 1, 0, delta, D_INNER, 0,
                dpb, nullptr, 0, 0, NT, D_INNER, DT_RANK, 2, 1);
    sel_scan<<<BB, D_INNER, 0, stream>>>(xz, xcs, dbc, delta, al, dp, ybuf);
    // X2 = X2 + ybuf(4096x1024) @ out_proj(1024x512)   (LDS kernel, residual in-place)
    launch_gemm(stream, ybuf, D_INNER, 1, 0, opw, DIM, 1, 0, X2, DIM, 0,
                nullptr, X2, DIM, 0, NT, DIM, D_INNER, 0, 1);
  }

  // ---- heads (tiny -> direct kernel)
  pool_mean<<<BB, 512, 0, stream>>>(X2, pooled);
  launch_gemm(stream, pooled, DIM, 1, 0, head_w1, DIM, 1, 0, h1, DIM, 0,
              head_b1, nullptr, 0, 0, BB, DIM, DIM, 1, 1);
  launch_gemm(stream, h1, DIM, 1, 0, head_w2, NCLS, 1, 0, out, NCLS, 0,
              head_b2, nullptr, 0, 0, BB, NCLS, DIM, 0, 1);
  launch_gemm(stream, pooled, DIM, 1, 0, reg_w1, DIM, 1, 0, r1, DIM, 0,
              reg_b1, nullptr, 0, 0, BB, DIM, DIM, 1, 1);
  launch_gemm(stream, r1, DIM, 1, 0, reg_w2, C_IN, 1, 0, out + BB * NCLS, C_IN, 0,
              reg_b2, nullptr, 0, 0, BB, C_IN, DIM, 0, 1);

  (void)in_sizes; (void)n_in; (void)out_size; (void)ws_size;
}